// LinearAttention_61503931678963
// MI455X (gfx1250) — compile-verified
//
#include <hip/hip_runtime.h>
#include <hip/hip_bf16.h>

typedef __attribute__((ext_vector_type(2))) float v2f;
typedef __attribute__((ext_vector_type(8))) float v8f;
typedef __attribute__((ext_vector_type(2))) int   v2i;

#define BATCH 16
#define DIM 256
#define HEADS 4
#define DHEAD 32
#define HIDDEN 128            // HEADS * DHEAD
#define OQKV 384              // 3 * HIDDEN
#define NPIX 4096             // 64*64
#define SCALE_Q 5.656854249492381f  // sqrt(32)

// ---- CDNA5 async load-to-LDS support (guarded; falls back to VGPR staging) ----
#if defined(__has_builtin)
#if __has_builtin(__builtin_amdgcn_global_load_async_to_lds_b64) && \
    __has_builtin(__builtin_amdgcn_global_load_async_to_lds_b32)
#define ASYNC_LDS 1
#endif
#endif
#ifndef ASYNC_LDS
#define ASYNC_LDS 0
#endif

#if ASYNC_LDS
__device__ __forceinline__ void async_copy_b64(const float* g, float* l) {
    __builtin_amdgcn_global_load_async_to_lds_b64(
        (__attribute__((address_space(1))) v2i*)g,
        (__attribute__((address_space(3))) v2i*)l, 0, 0);
}
__device__ __forceinline__ void async_copy_b32(const float* g, float* l) {
    __builtin_amdgcn_global_load_async_to_lds_b32(
        (__attribute__((address_space(1))) int*)g,
        (__attribute__((address_space(3))) int*)l, 0, 0);
}
__device__ __forceinline__ void wait_async0() {
#if __has_builtin(__builtin_amdgcn_s_wait_asynccnt)
    __builtin_amdgcn_s_wait_asynccnt(0);
#else
    asm volatile("s_wait_asynccnt 0x0" ::: "memory");
#endif
}
#endif

// ---------------------------------------------------------------------------
// Kernel 1: per-pixel RMS inverse norm over channels:  inv = 16 * rsqrt(sum x^2 + eps)
// ---------------------------------------------------------------------------
__global__ __launch_bounds__(256) void rms_in_kernel(const float* __restrict__ x,
                                                     float* __restrict__ inv) {
    int idx = blockIdx.x * 256 + threadIdx.x;      // [0, 16*4096)
    int b = idx >> 12, n = idx & 4095;
    const float* xb = x + (size_t)b * DIM * NPIX + n;
    float s = 0.f;
    for (int c = 0; c < DIM; ++c) {
        float v = xb[(size_t)c * NPIX];
        s += v * v;
    }
    inv[idx] = 16.0f * rsqrtf(s + 1e-12f);
}

// ---------------------------------------------------------------------------
// Kernel 2: QKV GEMM with fused RMS scaling.
// qkv[b][o][n] = sum_c w_qkv[o][c] * x[b][c][n] * inv[b][n] * g_in[c]
// 128(o) x 64(n) tile / block; K staged 32 at a time, double-buffered LDS.
// LDS uses k-pair-interleaved layout so each WMMA fragment is ONE ds_load_b64:
//   A: lA[(k>>1)*256 + o*2 + (k&1)],  B: lB[(k>>1)*128 + n*2 + (k&1)]
// ---------------------------------------------------------------------------
__global__ __launch_bounds__(256) void qkv_gemm(const float* __restrict__ x,
                                                const float* __restrict__ w,
                                                const float* __restrict__ g_in,
                                                const float* __restrict__ inv,
                                                float* __restrict__ qkv) {
    __shared__ __align__(16) float lA[2][16 * 256];  // 16 k-pairs x 128 o x 2
    __shared__ __align__(16) float lB[2][16 * 128];  // 16 k-pairs x 64 n x 2
    const int b  = blockIdx.z;
    const int o0 = blockIdx.x * 128;
    const int n0 = blockIdx.y * 64;
    const int tid  = threadIdx.x;
    const int wave = tid >> 5;
    const int lane = tid & 31;
    const int half = lane >> 4;      // 0: lanes 0-15 (K pair kk2), 1: lanes 16-31 (kk2+1)
    const int l16  = lane & 15;

    const float* xb   = x   + (size_t)b * DIM * NPIX;
    const float* invb = inv + (size_t)b * NPIX;

    v8f acc[4];
    for (int t = 0; t < 4; ++t)
        for (int j = 0; j < 8; ++j) acc[t][j] = 0.0f;

    auto loadA = [&](int buf, int k0) {
#if ASYNC_LDS
        // 2048 b64 chunks (one k-pair each), 8 per thread
        for (int i = 0; i < 8; ++i) {
            int id = tid + 256 * i;
            int r = id >> 4, j = id & 15;                 // o-row, k-pair
            async_copy_b64(w + (size_t)(o0 + r) * DIM + k0 + 2 * j,
                           &lA[buf][j * 256 + r * 2]);
        }
#else
        for (int i = tid; i < 128 * 32; i += 256) {
            int r = i >> 5, c = i & 31;
            lA[buf][(c >> 1) * 256 + r * 2 + (c & 1)] = w[(size_t)(o0 + r) * DIM + k0 + c];
        }
#endif
    };
    auto loadB = [&](int buf, int k0) {
        // fused scaling -> VGPR staging path
        for (int i = tid; i < 32 * 64; i += 256) {
            int r = i >> 6, c = i & 63;
            float v = xb[(size_t)(k0 + r) * NPIX + n0 + c];
            lB[buf][(r >> 1) * 128 + c * 2 + (r & 1)] = v * invb[n0 + c] * g_in[k0 + r];
        }
    };
    auto compute = [&](int buf) {
        const v2f* A2 = (const v2f*)lA[buf];
        const v2f* B2 = (const v2f*)lB[buf];
        const int oL = wave * 16 + l16;
        for (int kk2 = 0; kk2 < 16; kk2 += 2) {          // 8 K-steps of 4
            v2f a = A2[(kk2 + half) * 128 + oL];
            for (int nt = 0; nt < 4; ++nt) {
                v2f bb = B2[(kk2 + half) * 64 + nt * 16 + l16];
                acc[nt] = __builtin_amdgcn_wmma_f32_16x16x4_f32(
                    false, a, false, bb, (short)0, acc[nt], false, false);
            }
        }
    };

    loadA(0, 0); loadB(0, 0);
#if ASYNC_LDS
    wait_async0();
#endif
    __syncthreads();
    int buf = 0;
    for (int k0 = 0; k0 < DIM; k0 += 32) {
        if (k0 + 32 < DIM) { loadA(buf ^ 1, k0 + 32); loadB(buf ^ 1, k0 + 32); }
        compute(buf);
#if ASYNC_LDS
        wait_async0();
#endif
        __syncthreads();
        buf ^= 1;
    }

    // C/D layout: VGPR i -> (M=i, N=lane) lanes 0-15; (M=i+8, N=lane-16) lanes 16-31
    float* out = qkv + (size_t)b * OQKV * NPIX;
    for (int nt = 0; nt < 4; ++nt)
        for (int i = 0; i < 8; ++i) {
            int row = o0 + wave * 16 + i + half * 8;
            int col = n0 + nt * 16 + l16;
            out[(size_t)row * NPIX + col] = acc[nt][i];
        }
}

// ---------------------------------------------------------------------------
// Kernel 3: q softmax over the d axis (32 values per (b,h,n)), then / sqrt(32).
// ---------------------------------------------------------------------------
__global__ __launch_bounds__(256) void q_softmax_kernel(float* __restrict__ qkv) {
    int idx = blockIdx.x * 256 + threadIdx.x;   // [0, 16*4*4096)
    int b = idx >> 14;
    int r = idx & 16383;
    int h = r >> 12, n = r & 4095;
    float* base = qkv + ((size_t)b * OQKV + h * DHEAD) * NPIX + n;
    float m = -1e30f;
    for (int d = 0; d < DHEAD; ++d) m = fmaxf(m, base[(size_t)d * NPIX]);
    float s = 0.f;
    for (int d = 0; d < DHEAD; ++d) s += __expf(base[(size_t)d * NPIX] - m);
    float invs = 1.0f / (s * SCALE_Q);
    for (int d = 0; d < DHEAD; ++d)
        base[(size_t)d * NPIX] = __expf(base[(size_t)d * NPIX] - m) * invs;
}

// ---------------------------------------------------------------------------
// Kernel 4: per-row (b,h,d) max and sum-exp for k over n axis (4096 + 4 mem slots).
// ---------------------------------------------------------------------------
__global__ __launch_bounds__(256) void k_row_stats(const float* __restrict__ qkv,
                                                   const float* __restrict__ mem_kv,
                                                   float* __restrict__ kstat) {
    int rid = blockIdx.x;                 // [0, 16*128)
    int b = rid >> 7;
    int hd = rid & 127;
    const float* row = qkv + ((size_t)b * OQKV + HIDDEN + hd) * NPIX;
    const float* mk  = mem_kv + (size_t)hd * 4;   // mem_kv[0][h][d][m]
    __shared__ float red[256];
    int tid = threadIdx.x;
    float m = -1e30f;
    for (int n = tid; n < NPIX; n += 256) m = fmaxf(m, row[n]);
    if (tid < 4) m = fmaxf(m, mk[tid]);
    red[tid] = m; __syncthreads();
    for (int s = 128; s > 0; s >>= 1) {
        if (tid < s) red[tid] = fmaxf(red[tid], red[tid + s]);
        __syncthreads();
    }
    float rowmax = red[0]; __syncthreads();
    float sum = 0.f;
    for (int n = tid; n < NPIX; n += 256) sum += __expf(row[n] - rowmax);
    if (tid < 4) sum += __expf(mk[tid] - rowmax);
    red[tid] = sum; __syncthreads();
    for (int s = 128; s > 0; s >>= 1) {
        if (tid < s) red[tid] += red[tid + s];
        __syncthreads();
    }
    if (tid == 0) { kstat[rid * 2] = rowmax; kstat[rid * 2 + 1] = red[0]; }
}

// ---------------------------------------------------------------------------
// Kernel 5: context[b][h][d][e] = sum_n softmax_k[d,n] * v[e,n]  (n incl. mem)
// ---------------------------------------------------------------------------
__global__ __launch_bounds__(256) void context_kernel(const float* __restrict__ qkv,
                                                      const float* __restrict__ mem_kv,
                                                      const float* __restrict__ kstat,
                                                      float* __restrict__ ctx) {
    int b = blockIdx.x >> 2, h = blockIdx.x & 3;
    const float* krows = qkv + ((size_t)b * OQKV + HIDDEN + h * DHEAD) * NPIX;
    const float* vrows = qkv + ((size_t)b * OQKV + 2 * HIDDEN + h * DHEAD) * NPIX;
    const float* ks = kstat + ((size_t)b * HIDDEN + h * DHEAD) * 2;
    __shared__ float lk[DHEAD * 128];
    __shared__ float lv[DHEAD * 128];
    int tid = threadIdx.x;
    float acc[4]; int dd[4], ee[4];
    for (int i = 0; i < 4; ++i) {
        int p = tid + 256 * i; dd[i] = p >> 5; ee[i] = p & 31;
        float m = ks[dd[i] * 2], s = ks[dd[i] * 2 + 1];
        float a = 0.f;
        for (int mm = 0; mm < 4; ++mm) {
            float kw = __expf(mem_kv[(h * DHEAD + dd[i]) * 4 + mm] - m) / s;
            float vv = mem_kv[512 + (h * DHEAD + ee[i]) * 4 + mm];   // mem_kv[1]
            a += kw * vv;
        }
        acc[i] = a;
    }
    for (int n0 = 0; n0 < NPIX; n0 += 128) {
        for (int i = tid; i < DHEAD * 128; i += 256) {
            int r = i >> 7, c = i & 127;
            float m = ks[r * 2], s = ks[r * 2 + 1];
            lk[i] = __expf(krows[(size_t)r * NPIX + n0 + c] - m) / s;
            lv[i] = vrows[(size_t)r * NPIX + n0 + c];
        }
        __syncthreads();
        for (int i = 0; i < 4; ++i) {
            float a = acc[i];
            const float* kr = &lk[dd[i] * 128];
            const float* vr = &lv[ee[i] * 128];
            for (int c = 0; c < 128; ++c) a += kr[c] * vr[c];
            acc[i] = a;
        }
        __syncthreads();
    }
    float* cb = ctx + ((size_t)b * 4 + h) * 1024;
    for (int i = 0; i < 4; ++i) cb[dd[i] * 32 + ee[i]] = acc[i];
}

// ---------------------------------------------------------------------------
// Kernel 6: fold w_out with context:
// W2[b][o][h*32+d] = sum_e w_out[o][h*32+e] * ctx[b][h][d][e]
// ---------------------------------------------------------------------------
__global__ __launch_bounds__(256) void w2_kernel(const float* __restrict__ w_out,
                                                 const float* __restrict__ ctx,
                                                 float* __restrict__ W2) {
    int idx = blockIdx.x * 256 + threadIdx.x;   // [0, 16*256*128)
    int b = idx >> 15;
    int r = idx & 32767;
    int o = r >> 7;
    int hd = r & 127;
    int h = hd >> 5, d = hd & 31;
    const float* cb = ctx + ((size_t)b * 4 + h) * 1024 + d * 32;
    const float* wr = w_out + (size_t)o * HIDDEN + h * DHEAD;
    float a = 0.f;
    for (int e = 0; e < 32; ++e) a += wr[e] * cb[e];
    W2[idx] = a;
}

// ---------------------------------------------------------------------------
// Kernel 7: output GEMM: out[b][o][n] = sum_hd W2[b][o][hd]*q_sm[b][hd][n] + b_out[o]
// Same tiling; both A and B tiles are pure copies -> async-to-LDS eligible.
// ---------------------------------------------------------------------------
__global__ __launch_bounds__(256) void out_gemm(const float* __restrict__ W2,
                                                const float* __restrict__ qkv,
                                                const float* __restrict__ b_out,
                                                float* __restrict__ out) {
    __shared__ __align__(16) float lA[2][16 * 256];
    __shared__ __align__(16) float lB[2][16 * 128];
    const int b  = blockIdx.z;
    const int o0 = blockIdx.x * 128;
    const int n0 = blockIdx.y * 64;
    const int tid = threadIdx.x, wave = tid >> 5, lane = tid & 31;
    const int half = lane >> 4, l16 = lane & 15;
    const float* A = W2 + (size_t)b * DIM * HIDDEN;
    const float* B = qkv + (size_t)b * OQKV * NPIX;   // q rows are [0,128)

    v8f acc[4];
    for (int t = 0; t < 4; ++t)
        for (int j = 0; j < 8; ++j) acc[t][j] = 0.0f;

    auto loadA = [&](int buf, int k0) {
#if ASYNC_LDS
        for (int i = 0; i < 8; ++i) {
            int id = tid + 256 * i;
            int r = id >> 4, j = id & 15;
            async_copy_b64(A + (size_t)(o0 + r) * HIDDEN + k0 + 2 * j,
                           &lA[buf][j * 256 + r * 2]);
        }
#else
        for (int i = tid; i < 128 * 32; i += 256) {
            int r = i >> 5, c = i & 31;
            lA[buf][(c >> 1) * 256 + r * 2 + (c & 1)] = A[(size_t)(o0 + r) * HIDDEN + k0 + c];
        }
#endif
    };
    auto loadB = [&](int buf, int k0) {
#if ASYNC_LDS
        for (int i = 0; i < 8; ++i) {
            int id = tid + 256 * i;                    // 32x64 elements
            int r = id >> 6, c = id & 63;
            async_copy_b32(B + (size_t)(k0 + r) * NPIX + n0 + c,
                           &lB[buf][(r >> 1) * 128 + c * 2 + (r & 1)]);
        }
#else
        for (int i = tid; i < 32 * 64; i += 256) {
            int r = i >> 6, c = i & 63;
            lB[buf][(r >> 1) * 128 + c * 2 + (r & 1)] = B[(size_t)(k0 + r) * NPIX + n0 + c];
        }
#endif
    };
    auto compute = [&](int buf) {
        const v2f* A2 = (const v2f*)lA[buf];
        const v2f* B2 = (const v2f*)lB[buf];
        const int oL = wave * 16 + l16;
        for (int kk2 = 0; kk2 < 16; kk2 += 2) {
            v2f a = A2[(kk2 + half) * 128 + oL];
            for (int nt = 0; nt < 4; ++nt) {
                v2f bb = B2[(kk2 + half) * 64 + nt * 16 + l16];
                acc[nt] = __builtin_amdgcn_wmma_f32_16x16x4_f32(
                    false, a, false, bb, (short)0, acc[nt], false, false);
            }
        }
    };

    loadA(0, 0); loadB(0, 0);
#if ASYNC_LDS
    wait_async0();
#endif
    __syncthreads();
    int buf = 0;
    for (int k0 = 0; k0 < HIDDEN; k0 += 32) {
        if (k0 + 32 < HIDDEN) { loadA(buf ^ 1, k0 + 32); loadB(buf ^ 1, k0 + 32); }
        compute(buf);
#if ASYNC_LDS
        wait_async0();
#endif
        __syncthreads();
        buf ^= 1;
    }

    float* ob = out + (size_t)b * DIM * NPIX;
    for (int nt = 0; nt < 4; ++nt)
        for (int i = 0; i < 8; ++i) {
            int row = o0 + wave * 16 + i + half * 8;
            int col = n0 + nt * 16 + l16;
            ob[(size_t)row * NPIX + col] = acc[nt][i] + b_out[row];
        }
}

// ---------------------------------------------------------------------------
// Kernel 8: final RMS norm in place on d_out.
// ---------------------------------------------------------------------------
__global__ __launch_bounds__(256) void rms_out_kernel(float* __restrict__ out,
                                                      const float* __restrict__ g_out) {
    int idx = blockIdx.x * 256 + threadIdx.x;    // [0, 16*4096)
    int b = idx >> 12, n = idx & 4095;
    float* ob = out + (size_t)b * DIM * NPIX + n;
    float s = 0.f;
    for (int c = 0; c < DIM; ++c) {
        float v = ob[(size_t)c * NPIX];
        s += v * v;
    }
    float inv = 16.0f * rsqrtf(s + 1e-12f);
    for (int c = 0; c < DIM; ++c)
        ob[(size_t)c * NPIX] = ob[(size_t)c * NPIX] * inv * g_out[c];
}

// ---------------------------------------------------------------------------
extern "C" void kernel_launch(void* const* d_in, const int* in_sizes, int n_in,
                              void* d_out, int out_size, void* d_ws, size_t ws_size,
                              hipStream_t stream) {
    const float* x      = (const float*)d_in[0];   // [16][256][4096]
    const float* g_in   = (const float*)d_in[1];   // [256]
    const float* w_qkv  = (const float*)d_in[2];   // [384][256]
    const float* mem_kv = (const float*)d_in[3];   // [2][4][32][4]
    const float* w_out  = (const float*)d_in[4];   // [256][128]
    const float* b_out  = (const float*)d_in[5];   // [256]
    const float* g_out  = (const float*)d_in[6];   // [256]
    float* out = (float*)d_out;                    // [16][256][4096]

    float* ws = (float*)d_ws;
    float* inv   = ws;                                  // 65536
    float* qkv   = inv + (size_t)BATCH * NPIX;          // 16*384*4096
    float* kstat = qkv + (size_t)BATCH * OQKV * NPIX;   // 2048*2
    float* ctx   = kstat + (size_t)BATCH * HIDDEN * 2;  // 16*4*32*32
    float* W2    = ctx + (size_t)BATCH * HEADS * DHEAD * DHEAD; // 16*256*128

    rms_in_kernel  <<<BATCH * NPIX / 256, 256, 0, stream>>>(x, inv);
    qkv_gemm       <<<dim3(OQKV / 128, NPIX / 64, BATCH), 256, 0, stream>>>(x, w_qkv, g_in, inv, qkv);
    q_softmax_kernel<<<BATCH * HEADS * NPIX / 256, 256, 0, stream>>>(qkv);
    k_row_stats    <<<BATCH * HIDDEN, 256, 0, stream>>>(qkv, mem_kv, kstat);
    context_kernel <<<BATCH * HEADS, 256, 0, stream>>>(qkv, mem_kv, kstat, ctx);
    w2_kernel      <<<BATCH * DIM * HIDDEN / 256, 256, 0, stream>>>(w_out, ctx, W2);
    out_gemm       <<<dim3(DIM / 128, NPIX / 64, BATCH), 256, 0, stream>>>(W2, qkv, b_out, out);
    rms_out_kernel <<<BATCH * NPIX / 256, 256, 0, stream>>>(out, g_out);
}